// ConvChebTemp_64166811402347
// MI455X (gfx1250) — compile-verified
//
#include <hip/hip_runtime.h>
#include <hip/hip_bf16.h>

typedef __attribute__((ext_vector_type(2))) float v2f;
typedef __attribute__((ext_vector_type(8))) float v8f;
typedef int v4i __attribute__((vector_size(16)));
typedef __attribute__((address_space(1))) v4i* gv4i_p;   // global (AS1) 16B vector ptr
typedef __attribute__((address_space(3))) v4i* lv4i_p;   // LDS (AS3) 16B vector ptr

#define CV   12288   // V vertices
#define CB   16      // batch
#define CT   4       // time steps
#define CFIN 64
#define CFOUT 64
#define ROW  4096    // T*Fin*B columns per vertex row
#define KV   4
#define WR_ELEMS (4 * 256 * 32 * 2)   // [otile][ks][lane][j] reordered weights

// ---------------- CSR build (deterministic) ----------------

__global__ void zero_ints_kernel(int* __restrict__ p, int n) {
  int i = blockIdx.x * blockDim.x + threadIdx.x;
  if (i < n) p[i] = 0;
}

__global__ void count_rows_kernel(const int* __restrict__ rows, int* __restrict__ cnt, int nnz) {
  int e = blockIdx.x * blockDim.x + threadIdx.x;
  if (e < nnz) atomicAdd(&cnt[rows[e]], 1);
}

// single-block exclusive scan over V counts -> row_ptr[V+1]
__global__ __launch_bounds__(1024) void scan_rows_kernel(const int* __restrict__ cnt,
                                                         int* __restrict__ rp, int V) {
  __shared__ int sh[1024];
  __shared__ int carry;
  const int tid = threadIdx.x;
  if (tid == 0) { carry = 0; rp[0] = 0; }
  __syncthreads();
  for (int base = 0; base < V; base += 1024) {
    int v = base + tid;
    sh[tid] = (v < V) ? cnt[v] : 0;
    __syncthreads();
    for (int off = 1; off < 1024; off <<= 1) {
      int y = (tid >= off) ? sh[tid - off] : 0;
      __syncthreads();
      sh[tid] += y;
      __syncthreads();
    }
    if (v < V) rp[v + 1] = carry + sh[tid];
    __syncthreads();
    if (tid == 0) carry += sh[1023];
    __syncthreads();
  }
}

__global__ void fill_csr_kernel(const int* __restrict__ rows, const int* __restrict__ cols,
                                const float* __restrict__ vals, const int* __restrict__ rp,
                                int* __restrict__ cur, int* __restrict__ ccol,
                                int* __restrict__ ceid, float* __restrict__ cval, int nnz) {
  int e = blockIdx.x * blockDim.x + threadIdx.x;
  if (e >= nnz) return;
  int r = rows[e];
  int slot = rp[r] + atomicAdd(&cur[r], 1);
  ccol[slot] = cols[e];
  ceid[slot] = e;
  cval[slot] = vals[e];
}

// canonical per-row order (sort by original edge id) => bitwise-deterministic fp sums
__global__ void sort_rows_kernel(int* __restrict__ ccol, float* __restrict__ cval,
                                 int* __restrict__ ceid, const int* __restrict__ rp, int V) {
  int v = blockIdx.x * blockDim.x + threadIdx.x;
  if (v >= V) return;
  int s = rp[v], e = rp[v + 1];
  for (int i = s + 1; i < e; ++i) {
    int ke = ceid[i]; int kc = ccol[i]; float kv = cval[i];
    int j = i - 1;
    while (j >= s && ceid[j] > ke) {
      ceid[j + 1] = ceid[j]; ccol[j + 1] = ccol[j]; cval[j + 1] = cval[j];
      --j;
    }
    ceid[j + 1] = ke; ccol[j + 1] = kc; cval[j + 1] = kv;
  }
}

// ---------------- x0 = transpose(inputs,(1,2,3,0)).reshape(V, T*Fin*B) ----------------
// x0[v, (t*64+f)*16 + b] = inputs[b, v, t, f]
__global__ __launch_bounds__(256) void build_x0_kernel(const float* __restrict__ in,
                                                       float* __restrict__ x0) {
  __shared__ float tile[CB * 257];            // padded to dodge bank conflicts
  const int v = blockIdx.x;
  const int j = threadIdx.x;                   // 0..255 == t*64+f
#pragma unroll
  for (int b = 0; b < CB; ++b)
    tile[b * 257 + j] = in[((size_t)b * CV + v) * 256 + j];   // coalesced reads
  __syncthreads();
  float* __restrict__ xr = x0 + (size_t)v * ROW;
#pragma unroll
  for (int c = 0; c < 4; ++c) {
    int i = c * 256 + j;          // float4 index within the 4096-wide row
    int kap = i >> 2;             // = (4i)/16
    int b0 = (i & 3) * 4;         // = (4i)%16
    float4 w;
    w.x = tile[(b0 + 0) * 257 + kap];
    w.y = tile[(b0 + 1) * 257 + kap];
    w.z = tile[(b0 + 2) * 257 + kap];
    w.w = tile[(b0 + 3) * 257 + kap];
    ((float4*)xr)[i] = w;                                      // coalesced writes
  }
}

// ---------------- Wr[otile][ks=k*64+step][lane][j] = W[f_j, k, t_j, otile*16 + (lane&15)] --
// Makes the GEMM B-fragment load a single contiguous float2 per lane with a linear pointer.
__global__ __launch_bounds__(256) void build_wr_kernel(const float* __restrict__ W,
                                                       float* __restrict__ Wr) {
  int idx = blockIdx.x * blockDim.x + threadIdx.x;
  if (idx >= WR_ELEMS) return;
  int j    = idx & 1;
  int lane = (idx >> 1) & 31;
  int ks   = (idx >> 6) & 255;
  int ot   = idx >> 14;
  int k = ks >> 6, step = ks & 63;
  int kap = step * 4 + (lane >> 4) * 2 + j;    // K index inside 16x16x4 fragment
  int t = kap >> 6, f = kap & 63;
  Wr[idx] = W[((f * KV + k) * CT + t) * CFOUT + ot * 16 + (lane & 15)];
}

// ---------------- xout = alpha * (L @ xin) + beta * xm ----------------
__global__ __launch_bounds__(256) void spmm_cheb_kernel(const float* __restrict__ xin,
                                                        const float* __restrict__ xm,
                                                        float* __restrict__ xout,
                                                        const int* __restrict__ rp,
                                                        const int* __restrict__ ccol,
                                                        const float* __restrict__ cval,
                                                        float alpha, float beta) {
  const int v = blockIdx.x;
  const int c = blockIdx.y * 1024 + threadIdx.x * 4;
  const int s = rp[v], e = rp[v + 1];
  float4 acc = make_float4(0.f, 0.f, 0.f, 0.f);
  for (int j = s; j < e; ++j) {
    int col = ccol[j];
    float w = cval[j];
    if (j + 1 < e)   // uniform across block; hint next gathered row into cache
      __builtin_prefetch(xin + (size_t)ccol[j + 1] * ROW + c, 0, 1);
    float4 xr = *(const float4*)(xin + (size_t)col * ROW + c);
    acc.x += w * xr.x; acc.y += w * xr.y; acc.z += w * xr.z; acc.w += w * xr.w;
  }
  float4 r;
  if (beta != 0.0f) {
    float4 p = *(const float4*)(xm + (size_t)v * ROW + c);
    r.x = alpha * acc.x + beta * p.x;
    r.y = alpha * acc.y + beta * p.y;
    r.z = alpha * acc.z + beta * p.z;
    r.w = alpha * acc.w + beta * p.w;
  } else {
    r.x = alpha * acc.x; r.y = alpha * acc.y;
    r.z = alpha * acc.z; r.w = alpha * acc.w;
  }
  *(float4*)(xout + (size_t)v * ROW + c) = r;
}

// ---------------- out[b,v,o] = bias[o] + sum_{k,t,f} x_k[v,t,f,b] * W[f,k,t,o] ----------
// One block per vertex v; 4 waves; wave w owns o-tile [16w, 16w+16).
// 256 x v_wmma_f32_16x16x4_f32 per wave; A from LDS (async-staged), B from Wr (L2-hot).
__global__ __launch_bounds__(128) void gemm_out_kernel(const float* __restrict__ xall,
                                                       const float* __restrict__ Wr,
                                                       const float* __restrict__ bias,
                                                       float* __restrict__ out) {
  __shared__ __align__(16) float lx[KV * ROW];   // 64 KB: all 4 Chebyshev rows of vertex v
  const int v = blockIdx.x;
  const int tid = threadIdx.x;                   // 128

  // Stage x_0..x_3 rows into LDS via gfx1250 async copy (ASYNCcnt path), fallback: float4.
  for (int k = 0; k < KV; ++k) {
    const float4* src = (const float4*)(xall + ((size_t)k * CV + v) * ROW);
    float4* dst = (float4*)&lx[k * ROW];
    for (int i = tid; i < ROW / 4; i += 128) {
#if __has_builtin(__builtin_amdgcn_global_load_async_to_lds_b128)
      __builtin_amdgcn_global_load_async_to_lds_b128(
          (gv4i_p)(src + i), (lv4i_p)(dst + i), /*offset=*/0, /*cpol=*/0);
#else
      dst[i] = src[i];
#endif
    }
  }
#if __has_builtin(__builtin_amdgcn_global_load_async_to_lds_b128)
  asm volatile("s_wait_asynccnt 0x0" ::: "memory");
#endif
  __syncthreads();

  const int wave = tid >> 5;                   // o-tile id
  const int lane = tid & 31;
  const int half = lane >> 4;                  // 0: lanes 0-15, 1: lanes 16-31
  const int mn = lane & 15;                    // M (=b) for A, N (=o) for B/C

  v8f c;
  const float bo = bias[wave * 16 + mn];
#pragma unroll
  for (int r = 0; r < 8; ++r) c[r] = bo;       // C[m][n] = bias[o], broadcast over b

  // Fused ks = k*64+step loop: A LDS address is exactly linear (ks*64 + half*32 + mn),
  // B pointer advances 32 float2 per step -> minimal VALU per WMMA.
  const float* ap = lx + half * 32 + mn;
  const float2* bp = (const float2*)Wr + (size_t)wave * 256 * 32 + lane;
#pragma unroll 4
  for (int ks = 0; ks < 256; ++ks) {
    v2f a, b;
    a[0] = ap[0];
    a[1] = ap[16];
    float2 bw = *bp;
    b[0] = bw.x;
    b[1] = bw.y;
    c = __builtin_amdgcn_wmma_f32_16x16x4_f32(
        /*neg_a=*/false, a, /*neg_b=*/false, b,
        /*c_mod=*/(short)0, c, /*reuse_a=*/false, /*reuse_b=*/false);
    ap += 64;
    bp += 32;
  }

  // D layout: VGPR r -> M = r + 8*half, N = mn;  out[b, v, o]
#pragma unroll
  for (int r = 0; r < 8; ++r) {
    int bidx = r + half * 8;
    out[((size_t)bidx * CV + v) * CFOUT + wave * 16 + mn] = c[r];
  }
}

// ---------------- host ----------------

extern "C" void kernel_launch(void* const* d_in, const int* in_sizes, int n_in,
                              void* d_out, int out_size, void* d_ws, size_t ws_size,
                              hipStream_t stream) {
  const float* inputs  = (const float*)d_in[0];   // [B,V,T,Fin]
  const float* weight  = (const float*)d_in[1];   // [Fin,Kv,Kt,Fout]
  const float* bias    = (const float*)d_in[2];   // [Fout]
  const float* lapv    = (const float*)d_in[3];   // [NNZ]
  const int*   laprows = (const int*)d_in[4];
  const int*   lapcols = (const int*)d_in[5];
  const int nnz = in_sizes[3];

  const size_t xElems = (size_t)CV * ROW;
  float* x  = (float*)d_ws;                        // x0..x3 contiguous: 4 * 201 MB
  float* Wr = x + 4 * xElems;                      // 256 KB reordered weights (16B aligned)
  char* p = (char*)(Wr + WR_ELEMS);
  int* cnt = (int*)p;          p += (size_t)CV * sizeof(int);
  int* cur = (int*)p;          p += (size_t)CV * sizeof(int);
  int* rp  = (int*)p;          p += (size_t)(CV + 1) * sizeof(int);
  int* ccol = (int*)p;         p += (size_t)nnz * sizeof(int);
  int* ceid = (int*)p;         p += (size_t)nnz * sizeof(int);
  float* cval = (float*)p;     p += (size_t)nnz * sizeof(float);

  float* x0 = x;
  float* x1 = x + xElems;
  float* x2 = x + 2 * xElems;
  float* x3 = x + 3 * xElems;

  // 1) deterministic CSR build
  zero_ints_kernel<<<(2 * CV + 255) / 256, 256, 0, stream>>>(cnt, 2 * CV); // cnt & cur adjacent
  count_rows_kernel<<<(nnz + 255) / 256, 256, 0, stream>>>(laprows, cnt, nnz);
  scan_rows_kernel<<<1, 1024, 0, stream>>>(cnt, rp, CV);
  fill_csr_kernel<<<(nnz + 255) / 256, 256, 0, stream>>>(laprows, lapcols, lapv, rp, cur,
                                                         ccol, ceid, cval, nnz);
  sort_rows_kernel<<<(CV + 255) / 256, 256, 0, stream>>>(ccol, cval, ceid, rp, CV);

  // 2) layout transforms
  build_x0_kernel<<<CV, 256, 0, stream>>>(inputs, x0);
  build_wr_kernel<<<WR_ELEMS / 256, 256, 0, stream>>>(weight, Wr);

  // 3) Chebyshev recurrence (fused axpy)
  dim3 sg(CV, ROW / 1024);
  spmm_cheb_kernel<<<sg, 256, 0, stream>>>(x0, x0, x1, rp, ccol, cval, 1.0f, 0.0f);
  spmm_cheb_kernel<<<sg, 256, 0, stream>>>(x1, x0, x2, rp, ccol, cval, 2.0f, -1.0f);
  spmm_cheb_kernel<<<sg, 256, 0, stream>>>(x2, x1, x3, rp, ccol, cval, 2.0f, -1.0f);

  // 4) WMMA contraction + bias
  gemm_out_kernel<<<CV, 128, 0, stream>>>(x, Wr, bias, (float*)d_out);
}